// NnHalfKP_13580686590392
// MI455X (gfx1250) — compile-verified
//
#include <hip/hip_runtime.h>
#include <hip/hip_bf16.h>
#include <cstdint>
#include <cstddef>

#define B_SZ    512
#define FEATS   30
#define NNZ     (B_SZ * FEATS)
#define FT_OUT  512
#define N_FEAT  40960
#define N_VFEAT 640

typedef float v2f __attribute__((ext_vector_type(2)));
typedef float v8f __attribute__((ext_vector_type(8)));

// -------------------------------------------------------------------------
// Kernel 1: tiled transpose src[512, C] -> dst[C, 512]
// Stages 32x32 f32 tiles through LDS using the gfx1250 async global->LDS
// copy path (16 bytes per lane, tracked by ASYNCcnt).
// -------------------------------------------------------------------------
__global__ void nnue_transpose(const float* __restrict__ src,
                               float* __restrict__ dst, int C) {
  __shared__ float tile[32 * 36];            // stride 36 floats = 144B (16B aligned rows)
  const int c0   = blockIdx.x * 32;
  const int r0   = blockIdx.y * 32;
  const int t    = threadIdx.x;              // 0..255 (8 waves of 32)
  const int row  = t >> 3;                   // 0..31
  const int col4 = (t & 7) << 2;             // 0,4,...,28

  const float* g = src + (size_t)(r0 + row) * C + (c0 + col4);
  // LDS byte offset: low 32 bits of the flat shared address are the LDS offset.
  uint32_t lds_off = (uint32_t)(uintptr_t)(&tile[row * 36 + col4]);

  asm volatile("global_load_async_to_lds_b128 %0, %1, off"
               :: "v"(lds_off), "v"(g) : "memory");
  asm volatile("s_wait_asynccnt 0" ::: "memory");
  __syncthreads();

  // Transposed write-out: thread covers output row (c0+cc), elems r0+rr..r0+rr+3
  const int cc = row;
  const int rr = col4;
  float4 v;
  v.x = tile[(rr + 0) * 36 + cc];
  v.y = tile[(rr + 1) * 36 + cc];
  v.z = tile[(rr + 2) * 36 + cc];
  v.w = tile[(rr + 3) * 36 + cc];
  *(float4*)(dst + (size_t)(c0 + cc) * FT_OUT + (r0 + rr)) = v;
}

// -------------------------------------------------------------------------
// Kernel 2: embedding-bag gather + segment sum + bias + clamp.
// One 256-thread block per (batch row, perspective). Each thread owns a
// float2 slice of the 512-wide accumulator; every gathered row is a fully
// coalesced 2KB read from the (L2-resident) transposed tables.
// -------------------------------------------------------------------------
__global__ void nnue_gather(const int* __restrict__ stm_idx,
                            const int* __restrict__ nstm_idx,
                            const float* __restrict__ values,
                            const float* __restrict__ ft_t,
                            const float* __restrict__ fft_t,
                            const float* __restrict__ ft_b,
                            const float* __restrict__ fft_b,
                            float* __restrict__ hidden) {
  __shared__ int   s_col[FEATS];
  __shared__ float s_val[FEATS];
  const int b    = blockIdx.x >> 1;
  const int side = blockIdx.x & 1;           // 0 = stm, 1 = nstm (concat order)
  const int* idx = side ? nstm_idx : stm_idx;
  const int t    = threadIdx.x;              // 0..255

  if (t < FEATS) {
    s_col[t] = idx[NNZ + b * FEATS + t];     // row 1 of [2, NNZ] = cols
    s_val[t] = values[b * FEATS + t];
  }
  __syncthreads();

  const int j = t * 2;                       // this thread's float2 slot
  float2 acc = {0.f, 0.f};
#pragma unroll 5
  for (int k = 0; k < FEATS; ++k) {
    const int   c = s_col[k];
    const float v = s_val[k];
    const float2 a = *(const float2*)(ft_t  + (size_t)c * FT_OUT + j);
    const float2 g = *(const float2*)(fft_t + (size_t)(c % N_VFEAT) * FT_OUT + j);
    acc.x += v * (a.x + g.x);
    acc.y += v * (a.y + g.y);
  }

  float hx = acc.x + ft_b[j]     + fft_b[j];
  float hy = acc.y + ft_b[j + 1] + fft_b[j + 1];
  hx = fminf(fmaxf(hx, 0.f), 1.f);
  hy = fminf(fmaxf(hy, 0.f), 1.f);

  float2 h = {hx, hy};
  *(float2*)(hidden + (size_t)b * (2 * FT_OUT) + side * FT_OUT + j) = h;
}

// -------------------------------------------------------------------------
// Kernel 3: output layer  out[b] = sigmoid(hidden[b,:] . out_w + out_b)
// via V_WMMA_F32_16X16X4_F32. A = 16x4 hidden tile (documented A layout);
// B = out_w slice replicated across all 16 columns, so every column of D
// carries the same dot product and we read column 0 (lanes 0 and 16).
// One wave per 16-row tile; EXEC all-1s through the WMMA loop.
// -------------------------------------------------------------------------
__global__ void nnue_out(const float* __restrict__ hidden,
                         const float* __restrict__ out_w,
                         const float* __restrict__ out_b,
                         float* __restrict__ out) {
  const int lane = threadIdx.x;              // 0..31
  const int row0 = blockIdx.x * 16;
  const int m    = lane & 15;                // A-matrix row for this lane
  const int koff = (lane >> 4) * 2;          // half 0 -> K 0,1 ; half 1 -> K 2,3

  const float* hrow = hidden + (size_t)(row0 + m) * (2 * FT_OUT) + koff;
  const float* wrow = out_w + koff;

  v8f c = {0.f, 0.f, 0.f, 0.f, 0.f, 0.f, 0.f, 0.f};
  for (int k0 = 0; k0 < 2 * FT_OUT; k0 += 4) {
    v2f a  = *(const v2f*)(hrow + k0);       // A[m][koff..koff+1] of this K-chunk
    v2f bv = *(const v2f*)(wrow + k0);       // B[k][*] replicated across columns
    c = __builtin_amdgcn_wmma_f32_16x16x4_f32(false, a, false, bv,
                                              (short)0, c, false, false);
  }

  // D column 0: lane 0 holds rows 0..7 in c[0..7], lane 16 holds rows 8..15.
  if ((lane & 15) == 0) {
    const int   rbase = row0 + (lane >> 4) * 8;
    const float bb    = out_b[0];
#pragma unroll
    for (int r = 0; r < 8; ++r) {
      float x = c[r] + bb;
      out[rbase + r] = 1.0f / (1.0f + expf(-x));
    }
  }
}

// -------------------------------------------------------------------------
extern "C" void kernel_launch(void* const* d_in, const int* in_sizes, int n_in,
                              void* d_out, int out_size, void* d_ws, size_t ws_size,
                              hipStream_t stream) {
  (void)in_sizes; (void)n_in; (void)out_size; (void)ws_size;

  const int*   stm   = (const int*)  d_in[0];   // [2, NNZ]
  const int*   nstm  = (const int*)  d_in[1];   // [2, NNZ]
  const float* vals  = (const float*)d_in[2];   // [NNZ]
  const float* ft_w  = (const float*)d_in[3];   // [512, 40960]
  const float* ft_b  = (const float*)d_in[4];   // [512]
  const float* fft_w = (const float*)d_in[5];   // [512, 640]
  const float* fft_b = (const float*)d_in[6];   // [512]
  const float* out_w = (const float*)d_in[7];   // [1, 1024]
  const float* out_b = (const float*)d_in[8];   // [1]
  float*       out   = (float*)d_out;           // [512, 1]

  // Workspace layout (~87 MB): ft_t | fft_t | hidden
  float* ft_t   = (float*)d_ws;                                  // [40960, 512]
  float* fft_t  = ft_t  + (size_t)N_FEAT  * FT_OUT;              // [640, 512]
  float* hidden = fft_t + (size_t)N_VFEAT * FT_OUT;              // [512, 1024]

  nnue_transpose<<<dim3(N_FEAT  / 32, FT_OUT / 32), 256, 0, stream>>>(ft_w,  ft_t,  N_FEAT);
  nnue_transpose<<<dim3(N_VFEAT / 32, FT_OUT / 32), 256, 0, stream>>>(fft_w, fft_t, N_VFEAT);

  nnue_gather<<<dim3(2 * B_SZ), 256, 0, stream>>>(stm, nstm, vals, ft_t, fft_t,
                                                  ft_b, fft_b, hidden);

  nnue_out<<<dim3(B_SZ / 16), 32, 0, stream>>>(hidden, out_w, out_b, out);
}